// NeighborAggregator_89146341196441
// MI455X (gfx1250) — compile-verified
//
#include <hip/hip_runtime.h>
#include <math.h>

typedef __attribute__((ext_vector_type(2))) float v2f;
typedef __attribute__((ext_vector_type(8))) float v8f;

#define NROWS 8192
#define KP1   17      // K+1 gathered columns per row

// One wave computes A_raw for 16 rows via v_wmma_f32_16x16x4_f32 with an
// all-ones B matrix: D[m,n] = sum_k A[m,k] * 1.0  (exact f32 row-sum).
// 5 accumulating WMMAs cover K=0..19 (cols >= 17 contribute 0).
__global__ __launch_bounds__(256)
void rowsum_wmma_kernel(const float* __restrict__ data,
                        const int*   __restrict__ idx,
                        float*       __restrict__ araw) {
  const int lane       = threadIdx.x & 31;
  const int waveGlobal = blockIdx.x * 8 + (threadIdx.x >> 5);
  const int rowBase    = waveGlobal * 16;

  const int  r        = rowBase + (lane & 15);
  const int  jLane    = (lane >> 4) * 2;            // 0 for lanes 0-15, 2 for 16-31
  const long long rowOff = (long long)r * NROWS;    // row base into data
  const int  idxRow   = r * KP1;                    // row base into indices

  v8f acc = {};                                     // C starts at 0
  v2f bones = {1.0f, 1.0f};                         // all-ones B (layout-invariant)

#pragma unroll
  for (int c = 0; c < 5; ++c) {
    const int j0 = c * 4 + jLane;
    const int j1 = j0 + 1;
    const int j0c = (j0 < KP1) ? j0 : 0;            // clamp for safe load
    const int j1c = (j1 < KP1) ? j1 : 0;
    const int i0 = idx[idxRow + j0c];
    const int i1 = idx[idxRow + j1c];
    float a0 = data[rowOff + i0];
    float a1 = data[rowOff + i1];
    a0 = (j0 < KP1) ? a0 : 0.0f;                    // zero the padded K slots
    a1 = (j1 < KP1) ? a1 : 0.0f;
    v2f av = {a0, a1};
    // 8 args: (neg_a, A, neg_b, B, c_mod, C, reuse_a, reuse_b)
    acc = __builtin_amdgcn_wmma_f32_16x16x4_f32(false, av, false, bones,
                                                (short)0, acc, false, false);
  }

  // D layout (16x16 f32): VGPR v -> lane 0..15: M=v (N=lane); lane 16..31: M=v+8.
  // Column N=0 lives in lane 0 (rows 0..7) and lane 16 (rows 8..15).
  if (lane == 0) {
#pragma unroll
    for (int m = 0; m < 8; ++m) araw[rowBase + m] = acc[m];
  } else if (lane == 16) {
#pragma unroll
    for (int m = 0; m < 8; ++m) araw[rowBase + 8 + m] = acc[m];
  }
}

// Single-workgroup softmax over 8192 values: 1024 threads x 8 elements each.
__global__ __launch_bounds__(1024)
void softmax_kernel(const float* __restrict__ araw,
                    float*       __restrict__ alpha) {
  __shared__ float red[1024];
  const int tid = threadIdx.x;

  float vals[8];
  float m = -INFINITY;
#pragma unroll
  for (int i = 0; i < 8; ++i) {
    vals[i] = araw[tid + i * 1024];
    m = fmaxf(m, vals[i]);
  }
  red[tid] = m;
  __syncthreads();
  for (int s = 512; s > 0; s >>= 1) {
    if (tid < s) red[tid] = fmaxf(red[tid], red[tid + s]);
    __syncthreads();
  }
  m = red[0];
  __syncthreads();

  float sum = 0.0f;
#pragma unroll
  for (int i = 0; i < 8; ++i) {
    vals[i] = expf(vals[i] - m);
    sum += vals[i];
  }
  red[tid] = sum;
  __syncthreads();
  for (int s = 512; s > 0; s >>= 1) {
    if (tid < s) red[tid] += red[tid + s];
    __syncthreads();
  }
  const float inv = 1.0f / red[0];

#pragma unroll
  for (int i = 0; i < 8; ++i) alpha[tid + i * 1024] = vals[i] * inv;
}

extern "C" void kernel_launch(void* const* d_in, const int* in_sizes, int n_in,
                              void* d_out, int out_size, void* d_ws, size_t ws_size,
                              hipStream_t stream) {
  (void)in_sizes; (void)n_in; (void)out_size; (void)d_ws; (void)ws_size;
  const float* data = (const float*)d_in[0];
  const int*   idx  = (const int*)d_in[1];
  float* out   = (float*)d_out;
  float* alpha = out;            // first 8192 outputs
  float* araw  = out + NROWS;    // second 8192 outputs (also softmax input)

  // 8192 rows / 16 rows-per-wave = 512 waves = 64 blocks of 8 waves.
  rowsum_wmma_kernel<<<64, 256, 0, stream>>>(data, idx, araw);
  softmax_kernel<<<1, 1024, 0, stream>>>(araw, alpha);
}